// CausalSelfAttention_88579405513020
// MI455X (gfx1250) — compile-verified
//
#include <hip/hip_runtime.h>

typedef __attribute__((ext_vector_type(2))) float v2f;
typedef __attribute__((ext_vector_type(8))) float v8f;

#if defined(__HIP_DEVICE_COMPILE__) && !__has_builtin(__builtin_amdgcn_wmma_f32_16x16x4_f32)
#error "wmma_f32_16x16x4_f32 builtin not available on this toolchain (device pass)"
#endif

#define SEQ 512
#define DM  64
#define HD  16

// LDS layout (floats), padded strides for bank-conflict-free access
#define XS_N   (512 * 65)
#define QKV_N  (512 * 17)
#define W_N    (16 * 65)
#define WOC_N  (64 * 17)
#define PSC_N  (8 * 16 * 17)
#define SMEM_FLOATS (XS_N + 3 * QKV_N + 3 * W_N + WOC_N + PSC_N)
#define SMEM_BYTES  (SMEM_FLOATS * 4)

__global__ __launch_bounds__(256) void causal_attn_wmma_kernel(
    const float* __restrict__ x,
    const float* __restrict__ Wq, const float* __restrict__ Wk,
    const float* __restrict__ Wv, const float* __restrict__ Wo,
    float* __restrict__ out)
{
    extern __shared__ float smem[];
    float* Xs   = smem;                 // [512][65]  whole x tile for this (b,c)
    float* Qs   = Xs + XS_N;            // [512][17]  per-head Q
    float* Ks   = Qs + QKV_N;           // [512][17]
    float* Vs   = Ks + QKV_N;           // [512][17]
    float* Wqs  = Vs + QKV_N;           // [16][65]   head rows of Wq
    float* Wks  = Wqs + W_N;
    float* Wvs  = Wks + W_N;
    float* WoCs = Wvs + W_N;            // [64][17]   head columns of Wo
    float* Psc  = WoCs + WOC_N;         // [8][16*17] per-wave transpose scratch

    const int tid  = threadIdx.x;
    const int lane = tid & 31;
    const int wave = tid >> 5;
    const int hh   = lane >> 4;   // lane-half (selects K pair / upper rows)
    const int l16  = lane & 15;
    const int bc   = blockIdx.x >> 2;   // sequence index (0..255)
    const int h    = blockIdx.x & 3;    // head index

    // ---- Phase 0: cooperative loads into LDS ----
    const float4* xg4 = (const float4*)(x + (size_t)bc * SEQ * DM);
    for (int i = tid; i < SEQ * DM / 4; i += 256) {
        float4 f = xg4[i];
        int row = i >> 4, c = (i & 15) << 2;
        float* d = &Xs[row * 65 + c];
        d[0] = f.x; d[1] = f.y; d[2] = f.z; d[3] = f.w;
    }
    for (int i = tid; i < HD * DM; i += 256) {
        int r = i >> 6, e = i & 63;
        Wqs[r * 65 + e] = Wq[(h * HD + r) * DM + e];
        Wks[r * 65 + e] = Wk[(h * HD + r) * DM + e];
        Wvs[r * 65 + e] = Wv[(h * HD + r) * DM + e];
        int n = i >> 4, dd = i & 15;
        WoCs[n * 17 + dd] = Wo[n * DM + h * HD + dd];
    }
    __syncthreads();

    // ---- Phase 1: Q/K/V projections, y = x @ W_head^T, M=16 N=16 K=64 tiles ----
    for (int job = wave; job < 96; job += 8) {
        const int which = job >> 5;     // 0=Q 1=K 2=V
        const int st    = job & 31;     // s-tile
        const float* W  = (which == 0) ? Wqs : (which == 1) ? Wks : Wvs;
        float* dst      = (which == 0) ? Qs  : (which == 1) ? Ks  : Vs;
        v8f acc = {0.f,0.f,0.f,0.f,0.f,0.f,0.f,0.f};
#pragma unroll
        for (int c = 0; c < 16; ++c) {
            const int k0 = c * 4 + hh * 2;
            v2f a, b;
            a.x = Xs[(st * 16 + l16) * 65 + k0];       // A[m][k], m=l16
            a.y = Xs[(st * 16 + l16) * 65 + k0 + 1];
            b.x = W[l16 * 65 + k0];                    // B[k][n] = W[n][e]
            b.y = W[l16 * 65 + k0 + 1];
            acc = __builtin_amdgcn_wmma_f32_16x16x4_f32(
                false, a, false, b, (short)0, acc, false, false);
        }
#pragma unroll
        for (int v = 0; v < 8; ++v)
            dst[(st * 16 + v + 8 * hh) * 17 + l16] = acc[v];
    }
    __syncthreads();

    // ---- Phase 2: flash attention, one q-tile per wave iteration ----
    float* myP = Psc + wave * (16 * 17);
    for (int qt = wave; qt < 32; qt += 8) {
        v2f aq[4];
#pragma unroll
        for (int c = 0; c < 4; ++c) {
            const int k0 = c * 4 + hh * 2;
            aq[c].x = Qs[(qt * 16 + l16) * 17 + k0];
            aq[c].y = Qs[(qt * 16 + l16) * 17 + k0 + 1];
        }
        float m[8], lsum[8];
        v8f oacc = {0.f,0.f,0.f,0.f,0.f,0.f,0.f,0.f};
#pragma unroll
        for (int v = 0; v < 8; ++v) { m[v] = -__builtin_inff(); lsum[v] = 0.f; }

        for (int kt = 0; kt <= qt; ++kt) {
            // scores tile = Q_tile @ K_tile^T
            v8f s = {0.f,0.f,0.f,0.f,0.f,0.f,0.f,0.f};
#pragma unroll
            for (int c = 0; c < 4; ++c) {
                const int k0 = c * 4 + hh * 2;
                v2f b;
                b.x = Ks[(kt * 16 + l16) * 17 + k0];   // B[k][n] = K[key=n][d]
                b.y = Ks[(kt * 16 + l16) * 17 + k0 + 1];
                s = __builtin_amdgcn_wmma_f32_16x16x4_f32(
                    false, aq[c], false, b, (short)0, s, false, false);
            }
            // online softmax (row = v + 8*hh, cols across 16-lane half)
#pragma unroll
            for (int v = 0; v < 8; ++v) {
                float sv = s[v] * 0.25f;               // 1/sqrt(16)
                if (kt == qt && l16 > (v + 8 * hh)) sv = -__builtin_inff();
                float mx = sv;
#pragma unroll
                for (int off = 8; off > 0; off >>= 1)
                    mx = fmaxf(mx, __shfl_xor(mx, off, 16));
                float mn = fmaxf(m[v], mx);
                float p  = __expf(sv - mn);
                float rs = p;
#pragma unroll
                for (int off = 8; off > 0; off >>= 1)
                    rs += __shfl_xor(rs, off, 16);
                float corr = __expf(m[v] - mn);
                lsum[v] = lsum[v] * corr + rs;
                oacc[v] *= corr;
                m[v] = mn;
                myP[(v + 8 * hh) * 17 + l16] = p;      // C-layout -> scratch
            }
            // O += P @ V_tile  (re-read P in A layout; LDS is in-order per wave)
#pragma unroll
            for (int c = 0; c < 4; ++c) {
                const int k0 = c * 4 + hh * 2;
                v2f ap, bv;
                ap.x = myP[l16 * 17 + k0];
                ap.y = myP[l16 * 17 + k0 + 1];
                bv.x = Vs[(kt * 16 + k0) * 17 + l16];  // B[k][n] = V[key][dim=n]
                bv.y = Vs[(kt * 16 + k0 + 1) * 17 + l16];
                oacc = __builtin_amdgcn_wmma_f32_16x16x4_f32(
                    false, ap, false, bv, (short)0, oacc, false, false);
            }
        }
        // normalize and stage O for transpose
#pragma unroll
        for (int v = 0; v < 8; ++v) {
            oacc[v] /= lsum[v];
            myP[(v + 8 * hh) * 17 + l16] = oacc[v];
        }
        v2f ao[4];
#pragma unroll
        for (int c = 0; c < 4; ++c) {
            const int k0 = c * 4 + hh * 2;
            ao[c].x = myP[l16 * 17 + k0];
            ao[c].y = myP[l16 * 17 + k0 + 1];
        }
        // out_tile[16,64] += O[16,16] @ WoC^T ; merge heads via atomics
#pragma unroll
        for (int nt = 0; nt < 4; ++nt) {
            v8f co = {0.f,0.f,0.f,0.f,0.f,0.f,0.f,0.f};
#pragma unroll
            for (int c = 0; c < 4; ++c) {
                const int k0 = c * 4 + hh * 2;
                v2f bo;
                bo.x = WoCs[(nt * 16 + l16) * 17 + k0];
                bo.y = WoCs[(nt * 16 + l16) * 17 + k0 + 1];
                co = __builtin_amdgcn_wmma_f32_16x16x4_f32(
                    false, ao[c], false, bo, (short)0, co, false, false);
            }
#pragma unroll
            for (int v = 0; v < 8; ++v) {
                size_t oi = ((size_t)bc * SEQ + (size_t)(qt * 16 + v + 8 * hh)) * DM
                          + nt * 16 + l16;
                atomicAdd(&out[oi], co[v]);
            }
        }
    }
}

extern "C" void kernel_launch(void* const* d_in, const int* in_sizes, int n_in,
                              void* d_out, int out_size, void* d_ws, size_t ws_size,
                              hipStream_t stream) {
    (void)in_sizes; (void)n_in; (void)d_ws; (void)ws_size;
    const float* x  = (const float*)d_in[0];
    const float* Wq = (const float*)d_in[1];
    const float* Wk = (const float*)d_in[2];
    const float* Wv = (const float*)d_in[3];
    const float* Wo = (const float*)d_in[4];
    float* out = (float*)d_out;

    (void)hipMemsetAsync(d_out, 0, (size_t)out_size * sizeof(float), stream);
    (void)hipFuncSetAttribute((const void*)causal_attn_wmma_kernel,
                              hipFuncAttributeMaxDynamicSharedMemorySize, SMEM_BYTES);
    // one workgroup per (sequence, head): 256 seqs * 4 heads
    causal_attn_wmma_kernel<<<256 * 4, 256, SMEM_BYTES, stream>>>(
        x, Wq, Wk, Wv, Wo, out);
}